// VolumeRenderer_10024453669285
// MI455X (gfx1250) — compile-verified
//
#include <hip/hip_runtime.h>
#include <hip/hip_bf16.h>

#define N_RAYS    65536
#define N_PTS     128
#define EPSF      1e-10f
#define FAR_DELTA 1e10f
#define WPB       8          // waves per block (256 threads, wave32)

// ---------------------------------------------------------------------------
// Kernel 1: global max of depth_values. Depths are sorted ascending per ray,
// so max(depth) == max over rays of depth[r][127]. 65536 lanes, one per ray.
// ---------------------------------------------------------------------------
__global__ __launch_bounds__(256) void vr_max_kernel(
    const float* __restrict__ depth, unsigned int* __restrict__ maxbits) {
  const int r = blockIdx.x * blockDim.x + threadIdx.x;
  float v = depth[(size_t)r * N_PTS + (N_PTS - 1)];
#pragma unroll
  for (int off = 16; off >= 1; off >>= 1)
    v = fmaxf(v, __shfl_xor(v, off, 32));

  __shared__ float wmax[WPB];
  const int lane = threadIdx.x & 31;
  const int wave = threadIdx.x >> 5;
  if (lane == 0) wmax[wave] = v;
  __syncthreads();
  if (wave == 0) {
    float x = (lane < WPB) ? wmax[lane] : 0.0f;   // depths are positive
#pragma unroll
    for (int off = 4; off >= 1; off >>= 1)
      x = fmaxf(x, __shfl_xor(x, off, 32));
    if (lane == 0) atomicMax(maxbits, __float_as_uint(x));  // positive floats: uint order == float order
  }
}

// ---------------------------------------------------------------------------
// Kernel 2: one wave32 per ray, 4 samples per lane.
//  - rgb features async-staged into LDS (global_load_async_to_lds_b128),
//    overlapped with the transmittance scan; s_wait_asynccnt before use.
//  - exclusive cumsum of s = density*(delta+eps) via in-lane prefix +
//    wave shuffle scan (wave32).
//  - outputs reduced with a butterfly; lane 0 writes 3 rgb + 1 depth.
// ---------------------------------------------------------------------------
__global__ __launch_bounds__(256) void vr_render_kernel(
    const float* __restrict__ depth, const float* __restrict__ density,
    const float* __restrict__ feature, float* __restrict__ out,
    const unsigned int* __restrict__ maxbits) {
  __shared__ __align__(16) float feat_tile[WPB][32 * 12];

  const int lane = threadIdx.x & 31;
  const int wave = threadIdx.x >> 5;
  const int ray  = blockIdx.x * WPB + wave;

  // --- kick off async feature staging: 12 floats (48B) per lane ------------
  {
    // truncating a flat LDS pointer yields the workgroup-relative LDS offset
    unsigned int ldsa = (unsigned int)(uintptr_t)(&feat_tile[wave][lane * 12]);
    unsigned long long ga =
        (unsigned long long)(uintptr_t)(feature + (size_t)ray * (N_PTS * 3) + lane * 12);
    asm volatile("global_load_async_to_lds_b128 %0, %1, off"           :: "v"(ldsa), "v"(ga) : "memory");
    asm volatile("global_load_async_to_lds_b128 %0, %1, off offset:16" :: "v"(ldsa), "v"(ga) : "memory");
    asm volatile("global_load_async_to_lds_b128 %0, %1, off offset:32" :: "v"(ldsa), "v"(ga) : "memory");
  }

  // --- coalesced b128 loads of this lane's 4 depths / densities ------------
  const float4 dep = *(const float4*)(depth   + (size_t)ray * N_PTS + lane * 4);
  const float4 den = *(const float4*)(density + (size_t)ray * N_PTS + lane * 4);

  // deltas: neighbor depth from next lane; last sample gets FAR_DELTA
  const float nx = __shfl_down(dep.x, 1, 32);
  const float d0 = dep.y - dep.x;
  const float d1 = dep.z - dep.y;
  const float d2 = dep.w - dep.z;
  const float d3 = (lane == 31) ? FAR_DELTA : (nx - dep.w);

  // s = density * (delta + eps); in-lane exclusive prefix
  const float s0 = den.x * (d0 + EPSF);
  const float s1 = den.y * (d1 + EPSF);
  const float s2 = den.z * (d2 + EPSF);
  const float s3 = den.w * (d3 + EPSF);
  const float p1 = s0;
  const float p2 = p1 + s1;
  const float p3 = p2 + s2;
  const float tot = p3 + s3;

  // wave32 inclusive scan of per-lane totals -> exclusive prefix
  float inc = tot;
#pragma unroll
  for (int off = 1; off < 32; off <<= 1) {
    const float y = __shfl_up(inc, off, 32);
    if (lane >= off) inc += y;
  }
  const float excl = inc - tot;

  // transmittance and alpha weights (faithful: exp(-density*delta + eps))
  const float t0 = __expf(-excl);
  const float t1 = __expf(-(excl + p1));
  const float t2 = __expf(-(excl + p2));
  const float t3 = __expf(-(excl + p3));
  const float w0 = t0 * (1.0f - __expf(fmaf(-den.x, d0, EPSF)));
  const float w1 = t1 * (1.0f - __expf(fmaf(-den.y, d1, EPSF)));
  const float w2 = t2 * (1.0f - __expf(fmaf(-den.z, d2, EPSF)));
  const float w3 = t3 * (1.0f - __expf(fmaf(-den.w, d3, EPSF)));

  float dacc = w0 * dep.x + w1 * dep.y + w2 * dep.z + w3 * dep.w;

  // --- consume the async-staged features -----------------------------------
  asm volatile("s_wait_asynccnt 0" ::: "memory");
  const float4* tl = (const float4*)(&feat_tile[wave][lane * 12]);
  const float4 f0 = tl[0];
  const float4 f1 = tl[1];
  const float4 f2 = tl[2];

  float fx = w0 * f0.x + w1 * f0.w + w2 * f1.z + w3 * f2.y;
  float fy = w0 * f0.y + w1 * f1.x + w2 * f1.w + w3 * f2.z;
  float fz = w0 * f0.z + w1 * f1.y + w2 * f2.x + w3 * f2.w;

  // wave butterfly reduction of the four accumulators
#pragma unroll
  for (int off = 16; off >= 1; off >>= 1) {
    fx   += __shfl_xor(fx, off, 32);
    fy   += __shfl_xor(fy, off, 32);
    fz   += __shfl_xor(fz, off, 32);
    dacc += __shfl_xor(dacc, off, 32);
  }

  if (lane == 0) {
    const float inv = 1.0f / __uint_as_float(*maxbits);
    out[ray * 3 + 0]     = fx;
    out[ray * 3 + 1]     = fy;
    out[ray * 3 + 2]     = fz;
    out[N_RAYS * 3 + ray] = dacc * inv;   // sum(w*depth)/max == sum(w*depth/max)
  }
}

extern "C" void kernel_launch(void* const* d_in, const int* in_sizes, int n_in,
                              void* d_out, int out_size, void* d_ws, size_t ws_size,
                              hipStream_t stream) {
  (void)in_sizes; (void)n_in; (void)out_size; (void)ws_size;
  const float* depth   = (const float*)d_in[0];
  const float* density = (const float*)d_in[1];
  const float* feature = (const float*)d_in[2];
  float* out = (float*)d_out;
  unsigned int* maxbits = (unsigned int*)d_ws;

  // deterministic workspace init (graph-capture safe)
  hipMemsetAsync(d_ws, 0, sizeof(unsigned int), stream);

  vr_max_kernel<<<N_RAYS / 256, 256, 0, stream>>>(depth, maxbits);
  vr_render_kernel<<<N_RAYS / WPB, 256, 0, stream>>>(depth, density, feature, out, maxbits);
}